// SingleT2FLS_FWA_27530740367460
// MI455X (gfx1250) — compile-verified
//
#include <hip/hip_runtime.h>
#include <math.h>

// ---------------------------------------------------------------------------
// SingleT2FLS forward: interval type-2 fuzzy system, N samples, A=6, R=32.
// Wave32-per-sample, lane-per-rule design for MI455X (gfx1250).
// ---------------------------------------------------------------------------

#define RULES 32
#define ANTE  6
#define WAVES_PER_BLOCK 8
#define SPW   16                                   // samples per wave
#define SAMPLES_PER_BLOCK (WAVES_PER_BLOCK * SPW)  // 128
#define BLOCK_THREADS 256

// gfx1250 async global->LDS staging (ASYNCcnt-tracked), guarded so the host
// pass / older toolchains fall back to a synchronous copy.
#if defined(__HIP_DEVICE_COMPILE__) && defined(__gfx1250__) && \
    __has_builtin(__builtin_amdgcn_global_load_async_to_lds_b128) && \
    __has_builtin(__builtin_amdgcn_s_wait_asynccnt)
#define USE_ASYNC_LDS 1
#else
#define USE_ASYNC_LDS 0
#endif

#if USE_ASYNC_LDS
// Builtin signature (per hipcc diagnostic): pointee is a 4 x i32 vector.
typedef int async_v4i __attribute__((ext_vector_type(4)));
typedef __attribute__((address_space(1))) async_v4i* as1_v4i_ptr;
typedef __attribute__((address_space(3))) async_v4i* as3_v4i_ptr;
// Flat LDS addresses carry the LDS byte offset in the low 32 bits (ISA 10.2),
// so truncation through an integer is the correct generic->AS3 conversion.
#define TO_AS1(p) ((as1_v4i_ptr)(unsigned long long)(p))
#define TO_AS3(p) ((as3_v4i_ptr)(unsigned)(unsigned long long)(p))
#endif

// Branchless compare-exchange of a (key, u, l) triple pair, key ascending.
__device__ __forceinline__ void ce3(float& ca, float& ua, float& la,
                                    float& cb, float& ub, float& lb) {
  const bool sw = ca > cb;
  float tc = sw ? cb : ca; float tC = sw ? ca : cb;
  float tu = sw ? ub : ua; float tU = sw ? ua : ub;
  float tl = sw ? lb : la; float tL = sw ? la : lb;
  ca = tc; cb = tC; ua = tu; ub = tU; la = tl; lb = tL;
}

// Odd-even transposition sort of 6 triples by key (6 rounds, provably correct).
__device__ __forceinline__ void sort6(float* c, float* u, float* l) {
#define CE(i, j) ce3(c[i], u[i], l[i], c[j], u[j], l[j])
  CE(0,1); CE(2,3); CE(4,5);
  CE(1,2); CE(3,4);
  CE(0,1); CE(2,3); CE(4,5);
  CE(1,2); CE(3,4);
  CE(0,1); CE(2,3); CE(4,5);
  CE(1,2); CE(3,4);
#undef CE
}

// Karnik-Mendel left/right on 6 elements (per-lane, serial).
__device__ __forceinline__ float km6_left(const float* cin, const float* wu,
                                          const float* wl) {
  float c[6], u[6], l[6];
#pragma unroll
  for (int a = 0; a < 6; ++a) { c[a] = cin[a]; u[a] = wu[a]; l[a] = wl[a]; }
  sort6(c, u, l);
  float s0 = 0.f, s10 = 0.f;
#pragma unroll
  for (int a = 0; a < 6; ++a) { s0 = fmaf(c[a], l[a], s0); s10 += l[a]; }
  float best = __fdividef(s0, s10);
  float as = s0, a1 = s10;
#pragma unroll
  for (int a = 0; a < 6; ++a) {
    const float d = u[a] - l[a];
    as = fmaf(c[a], d, as);
    a1 += d;
    best = fminf(best, __fdividef(as, a1));
  }
  return best;
}

__device__ __forceinline__ float km6_right(const float* cin, const float* wu,
                                           const float* wl) {
  float c[6], u[6], l[6];
#pragma unroll
  for (int a = 0; a < 6; ++a) { c[a] = cin[a]; u[a] = wu[a]; l[a] = wl[a]; }
  sort6(c, u, l);
  float s0 = 0.f, s10 = 0.f;
#pragma unroll
  for (int a = 0; a < 6; ++a) { s0 = fmaf(c[a], u[a], s0); s10 += u[a]; }
  float best = __fdividef(s0, s10);
  float as = s0, a1 = s10;
#pragma unroll
  for (int a = 0; a < 6; ++a) {
    const float d = l[a] - u[a];
    as = fmaf(c[a], d, as);
    a1 += d;
    best = fmaxf(best, __fdividef(as, a1));
  }
  return best;
}

// Karnik-Mendel over 32 values distributed one-per-lane (wave32 collective).
// Bitonic sort of (c,u,l) triples by c across lanes, then prefix-sum ratios.
// EXEC must be full-wave at every shuffle (caller guarantees uniform guards).
template <bool LEFT>
__device__ __forceinline__ float wave_km(float c, float u, float l, int lane) {
#pragma unroll
  for (int k = 2; k <= 32; k <<= 1) {
#pragma unroll
    for (int j = k >> 1; j > 0; j >>= 1) {
      const float oc = __shfl_xor(c, j, 32);
      const float ou = __shfl_xor(u, j, 32);
      const float ol = __shfl_xor(l, j, 32);
      const bool lower = (lane & j) == 0;
      const bool asc = (lane & k) == 0;
      const bool keepMin = (asc == lower);
      const bool take = keepMin ? (oc < c) : (oc > c);
      c = take ? oc : c;
      u = take ? ou : u;
      l = take ? ol : l;
    }
  }
  const float bw = LEFT ? l : u;        // weight used by s0/s10
  const float d = LEFT ? (u - l) : (l - u);
  // totals: s0 = sum(c*bw), s10 = sum(bw)
  float s0 = c * bw, s10 = bw;
#pragma unroll
  for (int j = 1; j < 32; j <<= 1) {
    s0 += __shfl_xor(s0, j, 32);
    s10 += __shfl_xor(s10, j, 32);
  }
  // inclusive scans of c*d and d
  float sd = c * d, s1d = d;
#pragma unroll
  for (int j = 1; j < 32; j <<= 1) {
    const float t0 = __shfl_up(sd, (unsigned)j, 32);
    const float t1 = __shfl_up(s1d, (unsigned)j, 32);
    sd += (lane >= j) ? t0 : 0.0f;
    s1d += (lane >= j) ? t1 : 0.0f;
  }
  float best = __fdividef(s0 + sd, s10 + s1d);
  const float r0 = __fdividef(s0, s10);
  best = LEFT ? fminf(best, r0) : fmaxf(best, r0);
#pragma unroll
  for (int j = 1; j < 32; j <<= 1) {
    const float t = __shfl_xor(best, j, 32);
    best = LEFT ? fminf(best, t) : fmaxf(best, t);
  }
  return best;
}

__global__ __launch_bounds__(BLOCK_THREADS) void t2fls_fwd_kernel(
    const float* __restrict__ x, const float* __restrict__ FRB,
    const float* __restrict__ Wm, const float* __restrict__ Ws,
    const float* __restrict__ Bm, const float* __restrict__ Bs,
    float* __restrict__ out, int N) {
  __shared__ float ldsX[SAMPLES_PER_BLOCK * ANTE];  // 3 KB

  const int t = threadIdx.x;
  const int lane = t & 31;
  const int wave = t >> 5;
  const long long blockStart = (long long)blockIdx.x * SAMPLES_PER_BLOCK;

  // ---- Stage this block's input tile into LDS ----
#if USE_ASYNC_LDS
  if (blockStart + SAMPLES_PER_BLOCK <= (long long)N) {
    // 128 samples * 6 f32 = 3072 B = 192 x b128; threads 0..191 issue one each.
    if (t < (SAMPLES_PER_BLOCK * ANTE) / 4) {
      const float* g = x + blockStart * ANTE + t * 4;
      __builtin_amdgcn_global_load_async_to_lds_b128(TO_AS1(g),
                                                     TO_AS3(&ldsX[t * 4]),
                                                     /*offset=*/0, /*cpol=*/0);
    }
    __builtin_amdgcn_s_wait_asynccnt(0);
  } else {
    const int tot = (int)((long long)N - blockStart) * ANTE;
    for (int idx = t; idx < tot; idx += BLOCK_THREADS)
      ldsX[idx] = x[blockStart * ANTE + idx];
  }
#else
  {
    long long rem = (long long)N - blockStart;
    if (rem > SAMPLES_PER_BLOCK) rem = SAMPLES_PER_BLOCK;
    const int tot = (int)rem * ANTE;
    for (int idx = t; idx < tot; idx += BLOCK_THREADS)
      ldsX[idx] = x[blockStart * ANTE + idx];
  }
#endif
  __syncthreads();

  // ---- Per-lane rule constants (lane == rule), register resident ----
  const int r = lane;
  float m1[ANTE], m2[ANTE], md[ANTE], iv[ANTE], wu[ANTE], wl[ANTE];
#pragma unroll
  for (int a = 0; a < ANTE; ++a) {
    const float s = FRB[r * ANTE + a] + 1.0e-4f;
    const float p1 = FRB[r * ANTE + a + 1];
    const float p2 = FRB[r * ANTE + a + 2];
    const float lo = fminf(p1, p2), hi = fmaxf(p1, p2);
    m1[a] = lo;
    m2[a] = hi;
    md[a] = 0.5f * (lo + hi);
    iv[a] = __fdividef(1.0f, 2.0f * s * s);
    const float wm = Wm[r * ANTE + a], wsv = Ws[r * ANTE + a];
    wu[a] = wm + wsv;
    wl[a] = wm - wsv;
  }
  const float bm = Bm[r], bs = Bs[r];

  // ---- One sample per wave iteration ----
#pragma unroll 1
  for (int i = 0; i < SPW; ++i) {
    const long long sidx = blockStart + (long long)(wave * SPW + i);
    if (sidx >= (long long)N) break;  // wave-uniform guard: EXEC stays full
    const float* xs = &ldsX[(wave * SPW + i) * ANTE];  // broadcast LDS reads

    float U[ANTE], L[ANTE];
    float UU = 1.0f, LL = 1.0f;
#pragma unroll
    for (int a = 0; a < ANTE; ++a) {
      const float xv = xs[a];
      const float t1 = xv - m1[a];
      const float e1 = __expf(-(t1 * t1) * iv[a]);
      const float t2 = xv - m2[a];
      const float e2 = __expf(-(t2 * t2) * iv[a]);
      const float up = (xv < m1[a]) ? e1 : ((xv <= m2[a]) ? 1.0f : e2);
      const float lo = (xv <= md[a]) ? e2 : e1;
      U[a] = up;
      L[a] = lo;
      UU *= up;
      LL *= lo;
    }

    const float ys = km6_left(L, wu, wl) + bm + bs;   // y_small[n, r]
    const float yb = km6_right(U, wu, wl) + bm - bs;  // y_big[n, r]

    const float out_l = wave_km<true>(ys, UU, LL, lane);
    const float out_r = wave_km<false>(yb, UU, LL, lane);

    if (lane == 0) out[sidx] = 0.5f * (out_r + out_l);
  }
}

extern "C" void kernel_launch(void* const* d_in, const int* in_sizes, int n_in,
                              void* d_out, int out_size, void* d_ws,
                              size_t ws_size, hipStream_t stream) {
  (void)n_in;
  (void)d_ws;
  (void)ws_size;
  (void)out_size;
  const float* x = (const float*)d_in[0];    // (N, 6)
  const float* FRB = (const float*)d_in[1];  // (R*A*3,)
  const float* Wm = (const float*)d_in[2];   // (R, A)
  const float* Ws = (const float*)d_in[3];   // (R, A)
  const float* Bm = (const float*)d_in[4];   // (R,)
  const float* Bs = (const float*)d_in[5];   // (R,)
  float* out = (float*)d_out;                // (N,)

  const int N = in_sizes[0] / ANTE;
  const int nBlocks = (N + SAMPLES_PER_BLOCK - 1) / SAMPLES_PER_BLOCK;
  t2fls_fwd_kernel<<<nBlocks, BLOCK_THREADS, 0, stream>>>(x, FRB, Wm, Ws, Bm,
                                                          Bs, out, N);
}